// MultiBoxLoss_53145925320881
// MI455X (gfx1250) — compile-verified
//
#include <hip/hip_runtime.h>
#include <stdint.h>

#define NN 128
#define BB 21824

typedef unsigned int __attribute__((ext_vector_type(4))) u32x4;
typedef int          __attribute__((ext_vector_type(8))) i32x8;
typedef int          __attribute__((ext_vector_type(4))) i32x4;
typedef float        __attribute__((ext_vector_type(2))) v2f;
typedef float        __attribute__((ext_vector_type(8))) v8f;

__device__ __forceinline__ float smooth_l1(float d) {
    float a = fabsf(d);
    return a < 1.0f ? 0.5f * d * d : a - 0.5f;
}

__device__ __forceinline__ float wave_sum_f(float v) {
    #pragma unroll
    for (int m = 16; m >= 1; m >>= 1) v += __shfl_xor(v, m, 32);
    return v;
}
__device__ __forceinline__ int wave_sum_i(int v) {
    #pragma unroll
    for (int m = 16; m >= 1; m >>= 1) v += __shfl_xor(v, m, 32);
    return v;
}

// ---------------- kernel 0: zero the workspace accumulators ----------------
__global__ void k_zero(float* acc, int* npos) {
    for (int i = threadIdx.x; i < 640; i += 256) acc[i] = 0.0f;
    for (int i = threadIdx.x; i < NN; i += 256) npos[i] = 0;
}

// ---------------- kernel 1: elementwise masked sums ----------------
// acc[0..127]   = per-row smooth-L1 loc sum (positives only)
// acc[128..255] = per-row smooth-L1 landmark sum (positives only)
// acc[256..383] = per-row CE sum over positive anchors
// npos[0..127]  = per-row positive count
__global__ void __launch_bounds__(256)
k_elem(const float* __restrict__ locP, const float* __restrict__ locT,
       const float* __restrict__ confP, const int* __restrict__ confT,
       const float* __restrict__ lmkP, const float* __restrict__ lmkT,
       float* __restrict__ acc, int* __restrict__ npos) {
    int n = blockIdx.y;
    int b = blockIdx.x * blockDim.x + threadIdx.x;

    float ls = 0.0f, ms = 0.0f, cs = 0.0f;
    int np = 0;
    if (b < BB) {
        size_t base = (size_t)n * BB + b;
        int t = confT[base];
        if (t > 0) {
            np = 1;
            // loc: 4 floats, 16B aligned
            float4 p = *(const float4*)(locP + base * 4);
            float4 q = *(const float4*)(locT + base * 4);
            ls = smooth_l1(p.x - q.x) + smooth_l1(p.y - q.y) +
                 smooth_l1(p.z - q.z) + smooth_l1(p.w - q.w);
            // landmarks: 10 floats, 8B aligned -> 5x float2
            const float2* mp = (const float2*)(lmkP + base * 10);
            const float2* mt = (const float2*)(lmkT + base * 10);
            #pragma unroll
            for (int j = 0; j < 5; ++j) {
                float2 a = mp[j], c = mt[j];
                ms += smooth_l1(a.x - c.x) + smooth_l1(a.y - c.y);
            }
            // CE for positive anchor
            float2 cp = *(const float2*)(confP + base * 2);
            float l0 = cp.x, l1 = cp.y;
            float mx = fmaxf(l0, l1);
            float lse = mx + logf(expf(l0 - mx) + expf(l1 - mx));
            cs = lse - l1;   // t > 0 => target class 1
        }
    }

    ls = wave_sum_f(ls);
    ms = wave_sum_f(ms);
    cs = wave_sum_f(cs);
    np = wave_sum_i(np);

    __shared__ float sl[8], sm[8], sc[8];
    __shared__ int   sn[8];
    int wid = threadIdx.x >> 5, lane = threadIdx.x & 31;
    if (lane == 0) { sl[wid] = ls; sm[wid] = ms; sc[wid] = cs; sn[wid] = np; }
    __syncthreads();
    if (threadIdx.x == 0) {
        float L = 0, M = 0, C = 0; int P = 0;
        #pragma unroll
        for (int i = 0; i < 8; ++i) { L += sl[i]; M += sm[i]; C += sc[i]; P += sn[i]; }
        atomicAdd(&acc[n], L);
        atomicAdd(&acc[128 + n], M);
        atomicAdd(&acc[256 + n], C);
        atomicAdd(&npos[n], P);
    }
}

// ---------------- kernel 2: per-row hard-negative mining ----------------
// One workgroup per row. conf_targets row is staged into LDS via the Tensor
// Data Mover, CE (cl) bit patterns live in LDS, 4x8-bit MSB radix select
// finds the k-th largest value T exactly; sum of top-k = sum(cl > T) + kk*T.
// acc[384..511] = per-row negative CE sum ; acc[512..639] = float(num_pos)
__global__ void __launch_bounds__(1024)
k_select(const float* __restrict__ confP, const int* __restrict__ confT,
         float* __restrict__ acc, const int* __restrict__ npos) {
    extern __shared__ unsigned smem[];
    unsigned* clbuf = smem;                 // BB words: targets, then cl bits
    int* hist = (int*)(smem + BB);          // 256 bins

    __shared__ unsigned s_prefix;
    __shared__ int s_kk;
    __shared__ float s_red[32];

    int n = blockIdx.x;
    int tid = threadIdx.x;

    // --- TDM: DMA the conf_targets row (BB x int32) into LDS (wave 0 only) ---
    if (tid < 32) {
        unsigned lds_addr = (unsigned)(uintptr_t)clbuf;
        uint64_t ga = (uint64_t)(uintptr_t)(confT + (size_t)n * BB);
        u32x4 g0;
        g0.x = 1u;                                              // count=1, user mode
        g0.y = lds_addr;                                        // LDS dest (bytes)
        g0.z = (unsigned)(ga & 0xFFFFFFFFu);                    // global_addr[31:0]
        g0.w = (unsigned)((ga >> 32) & 0x01FFFFFFu) | (2u << 30); // addr[56:32] | type=2
        i32x8 g1;
        g1[0] = (2 << 16);                                      // data_size = 4B
        g1[1] = (int)((BB & 0xFFFF) << 16);                     // tensor_dim0 lo16
        g1[2] = (int)(((BB >> 16) & 0xFFFF) | (1 << 16));       // tensor_dim0 hi | tensor_dim1=1
        g1[3] = (int)((BB & 0xFFFF) << 16);                     // tile_dim0 = BB
        g1[4] = 1;                                              // tile_dim1 = 1, tile_dim2 = 0
        g1[5] = BB;                                             // tensor_dim0_stride lo32
        g1[6] = 0;
        g1[7] = 0;
        i32x4 z4 = {0, 0, 0, 0};
        i32x8 z8 = {0, 0, 0, 0, 0, 0, 0, 0};
        __builtin_amdgcn_tensor_load_to_lds(g0, g1, z4, z4, z8, 0);
        __builtin_amdgcn_s_wait_tensorcnt(0);
    }
    __syncthreads();

    int np = npos[n];
    int k = 3 * np;
    if (k > BB - 1) k = BB - 1;

    // --- phase 0: targets(LDS) + conf_preds(global) -> cl bits in LDS in-place ---
    for (int b = tid; b < BB; b += 1024) {
        int t = (int)clbuf[b];
        float2 cp = *(const float2*)(confP + ((size_t)n * BB + b) * 2);
        float l0 = cp.x, l1 = cp.y;
        float mx = fmaxf(l0, l1);
        float lse = mx + logf(expf(l0 - mx) + expf(l1 - mx));
        float ce = lse - (t > 0 ? l1 : l0);
        float cl = (t > 0) ? 0.0f : ce;      // cl >= 0 always
        clbuf[b] = __float_as_uint(cl);
    }
    __syncthreads();

    float negsum = 0.0f;
    if (k > 0) {
        if (tid == 0) { s_prefix = 0u; s_kk = k; }
        __syncthreads();
        // --- 4-pass MSB radix select for the k-th largest cl ---
        for (int pass = 3; pass >= 0; --pass) {
            for (int i = tid; i < 256; i += 1024) hist[i] = 0;
            __syncthreads();
            unsigned prefix = s_prefix;
            int shift = pass * 8;
            unsigned mask = (pass == 3) ? 0u : (0xFFFFFFFFu << (shift + 8));
            for (int b = tid; b < BB; b += 1024) {
                unsigned v = clbuf[b];
                if ((v & mask) == prefix) atomicAdd(&hist[(v >> shift) & 255], 1);
            }
            __syncthreads();
            if (tid == 0) {
                int kk = s_kk, cum = 0, d = 255;
                for (; d >= 0; --d) {
                    int h = hist[d];
                    if (cum + h >= kk) break;
                    cum += h;
                }
                if (d < 0) d = 0;
                s_prefix = prefix | ((unsigned)d << shift);
                s_kk = kk - cum;
            }
            __syncthreads();
        }
        unsigned T = s_prefix;
        int kk = s_kk;                        // multiplicity of T to include
        float sum = 0.0f;
        for (int b = tid; b < BB; b += 1024) {
            unsigned v = clbuf[b];
            if (v > T) sum += __uint_as_float(v);   // uint cmp == float cmp (cl>=0)
        }
        sum = wave_sum_f(sum);
        int wid = tid >> 5, lane = tid & 31;
        if (lane == 0) s_red[wid] = sum;
        __syncthreads();
        if (tid < 32) {
            float v = s_red[tid];
            v = wave_sum_f(v);
            if (tid == 0) negsum = v + (float)kk * __uint_as_float(T);
        }
    }
    if (tid == 0) {
        acc[384 + n] = negsum;
        acc[512 + n] = (float)np;
    }
}

// ---------------- kernel 3: WMMA final reduction ----------------
// numerator = sum(acc[0..511]) = loc + lmk + ce_pos + ce_neg (x4/10 factors cancel)
// denominator = sum(acc[512..639]) = num_matched ; out = num/den
__global__ void __launch_bounds__(32) k_final(const float* __restrict__ acc,
                                              float* __restrict__ out) {
    int lane = threadIdx.x;
    v2f ones; ones.x = 1.0f; ones.y = 1.0f;     // A = all-ones 16x4
    v8f cnum = {}; v8f cden = {};
    #pragma unroll
    for (int c = 0; c < 8; ++c) {               // 8 x 64 = 512 numerator partials
        v2f b;
        b.x = acc[c * 64 + lane];
        b.y = acc[c * 64 + 32 + lane];
        cnum = __builtin_amdgcn_wmma_f32_16x16x4_f32(false, ones, false, b,
                                                     (short)0, cnum, false, false);
    }
    #pragma unroll
    for (int c = 0; c < 2; ++c) {               // 2 x 64 = 128 num_pos partials
        v2f b;
        b.x = acc[512 + c * 64 + lane];
        b.y = acc[512 + c * 64 + 32 + lane];
        cden = __builtin_amdgcn_wmma_f32_16x16x4_f32(false, ones, false, b,
                                                     (short)0, cden, false, false);
    }
    // D[m][n] = sum_k B[k][n]; summing VGPR0 over all 32 lanes gives 2*total
    // for both num and den -> factor cancels in the ratio.
    float num = cnum[0], den = cden[0];
    #pragma unroll
    for (int m = 16; m >= 1; m >>= 1) {
        num += __shfl_xor(num, m, 32);
        den += __shfl_xor(den, m, 32);
    }
    if (lane == 0) out[0] = num / den;
}

extern "C" void kernel_launch(void* const* d_in, const int* in_sizes, int n_in,
                              void* d_out, int out_size, void* d_ws, size_t ws_size,
                              hipStream_t stream) {
    const float* locP  = (const float*)d_in[0];
    const float* locT  = (const float*)d_in[1];
    const float* confP = (const float*)d_in[2];
    const int*   confT = (const int*)d_in[3];
    const float* lmkP  = (const float*)d_in[4];
    const float* lmkT  = (const float*)d_in[5];

    float* acc = (float*)d_ws;       // 640 floats of accumulators
    int* npos  = (int*)d_ws + 640;   // 128 ints

    k_zero<<<1, 256, 0, stream>>>(acc, npos);

    dim3 g1((BB + 255) / 256, NN);
    k_elem<<<g1, 256, 0, stream>>>(locP, locT, confP, confT, lmkP, lmkT, acc, npos);

    size_t smem = (size_t)BB * sizeof(unsigned) + 256 * sizeof(int);
    k_select<<<NN, 1024, smem, stream>>>(confP, confT, acc, npos);

    k_final<<<1, 32, 0, stream>>>(acc, (float*)d_out);
}